// KMeans_84482006712834
// MI455X (gfx1250) — compile-verified
//
#include <hip/hip_runtime.h>
#include <cstdint>
#include <cstddef>

// Problem constants (from reference): x [4,3,384,384] f32, K=32, 50 iters.
#define BATCH 4
#define KC 32
#define HH 384
#define WW 384
#define NPIX (HH * WW)                  // 147456 points per batch
#define D8 8                            // D=5 padded to 8 (dim5 = 1.0 homog. term)
#define TILES_PER_BATCH (NPIX / 16)     // 9216 16-point tiles
#define WAVES_PER_BLOCK 8               // 256 threads = 8 wave32
#define TILES_PER_WAVE 4
#define ASSIGN_BLOCKS_X (TILES_PER_BATCH / (WAVES_PER_BLOCK * TILES_PER_WAVE)) // 288
#define GS_STRIDE (KC * 6)              // per-batch accumulator: 5 sums + count
#define SWAPX16 0x401F                  // ds_swizzle group-of-32: xor=0x10, and=0x1F

typedef __attribute__((ext_vector_type(2))) float v2f;
typedef __attribute__((ext_vector_type(8))) float v8f;

__device__ __forceinline__ v8f wmma4(v2f a, v2f b, v8f c) {
    return __builtin_amdgcn_wmma_f32_16x16x4_f32(false, a, false, b, (short)0, c,
                                                 false, false);
}

// ---------------------------------------------------------------------------
// Kernel 1: padded feature rows [B][N][8] = {ch0,ch1,ch2,yg,xg, 1, 0, 0}
// dim5 = 1.0 is the homogeneous term that picks up ||c||^2 from the centroid
// operand, so the WMMA output is the distance directly.
// 18.9 MB -> resident in the 192 MB L2 for all 51 assignment passes.
// ---------------------------------------------------------------------------
__global__ void feat_kernel(const float* __restrict__ x, float* __restrict__ feats) {
    int p = blockIdx.x * blockDim.x + threadIdx.x;
    int b = blockIdx.y;
    if (p >= NPIX) return;
    const float sc = 2.0f / 255.0f;
    size_t base = (size_t)b * 3 * NPIX + p;
    float f0 = x[base] * sc - 1.0f;
    float f1 = x[base + NPIX] * sc - 1.0f;
    float f2 = x[base + 2 * (size_t)NPIX] * sc - 1.0f;
    int h = p / WW, w = p % WW;
    float yg = 2.0f * (float)h / (float)(HH - 1) - 1.0f;
    float xg = 2.0f * (float)w / (float)(WW - 1) - 1.0f;
    float4* out = (float4*)(feats + ((size_t)b * NPIX + p) * D8);
    out[0] = make_float4(f0, f1, f2, yg);
    out[1] = make_float4(xg, 1.0f, 0.0f, 0.0f);
}

// ---------------------------------------------------------------------------
// Kernel 2: threefry-2x32 (key = jax.random.key(42) -> {0,42}) to pick the
// K initial centroid indices per batch; writes centroid OPERAND form
// cop = {-2c0..-2c4, ||c||^2, 0, 0}; zeroes accumulators.
// ---------------------------------------------------------------------------
__device__ __forceinline__ unsigned rotl32(unsigned v, int r) {
    return (v << r) | (v >> (32 - r));
}

__global__ void init_kernel(const float* __restrict__ feats,
                            float* __restrict__ cop,
                            float* __restrict__ gsums) {
    int i = threadIdx.x;                 // 0..127 == flat (b,k)
    for (int j = 0; j < 6; ++j) gsums[i * 6 + j] = 0.0f;

    // jax random_bits(128): counters split into halves (i, i+64)
    unsigned x0 = (i < 64) ? (unsigned)i : (unsigned)(i - 64);
    unsigned x1 = (i < 64) ? (unsigned)(i + 64) : (unsigned)i;
    const unsigned k0 = 0u, k1 = 42u;
    unsigned ks[3] = { k0, k1, 0x1BD11BDAu ^ k0 ^ k1 };
    const int R0[4] = { 13, 15, 26, 6 };
    const int R1[4] = { 17, 29, 16, 24 };
    x0 += ks[0]; x1 += ks[1];
    #pragma unroll
    for (int g = 0; g < 5; ++g) {
        const int* R = (g & 1) ? R1 : R0;
        #pragma unroll
        for (int j = 0; j < 4; ++j) { x0 += x1; x1 = rotl32(x1, R[j]); x1 ^= x0; }
        x0 += ks[(g + 1) % 3];
        x1 += ks[(g + 2) % 3] + (unsigned)(g + 1);
    }
    unsigned bits = (i < 64) ? x0 : x1;
    const unsigned span = NPIX;
    unsigned mult = 65536u % span;             // jax randint: (hi*mult + lo) % span
    unsigned hiб = bits >> 16, lo = bits & 0xFFFFu;
    unsigned idx = (unsigned)(((unsigned long long)hiб * mult + lo) % span);

    int b = i / KC;
    const float* F = feats + ((size_t)b * NPIX + idx) * D8;
    float nrm = 0.0f;
    float* cp = cop + (size_t)i * D8;
    #pragma unroll
    for (int j = 0; j < 5; ++j) {
        float v = F[j];
        cp[j] = -2.0f * v;
        nrm += v * v;
    }
    cp[5] = nrm; cp[6] = 0.0f; cp[7] = 0.0f;
}

// ---------------------------------------------------------------------------
// Kernel 3: fused assignment + accumulation (A = centroid operands, B = points).
// D layout: VGPR r, lane l -> cluster r + (l>>4)*8, point (l&15).
// => each lane holds 16 of 32 distances for ITS point: register-local argmin
//    scan + one SWAPX16 ds_swizzle merges the halves. No bpermute, no branches.
// ---------------------------------------------------------------------------
__global__ void assign_kernel(const float* __restrict__ feats,
                              const float* __restrict__ cop,
                              float* __restrict__ gsums,
                              int* __restrict__ labels) {
    __shared__ float smem[GS_STRIDE];
    int tid = threadIdx.x;
    if (tid < GS_STRIDE) smem[tid] = 0.0f;
    __syncthreads();

    int lane = tid & 31;
    int wave = tid >> 5;
    int b = blockIdx.y;
    int hi = lane >> 4;          // half-wave id
    int ln = lane & 15;
    int kp = hi * 2;             // K-pair within 4-chunk

    // A tiles (iteration-invariant across the point loop):
    // lane holds cluster row ln, K-pair kp..kp+1
    const float* C0 = cop + ((size_t)b * KC + ln) * D8;   // clusters 0..15
    const float* C1 = C0 + 16 * D8;                       // clusters 16..31
    v2f a0_lo = *(const v2f*)(C0 + kp);
    v2f a0_hi = *(const v2f*)(C0 + 4 + kp);
    v2f a1_lo = *(const v2f*)(C1 + kp);
    v2f a1_hi = *(const v2f*)(C1 + 4 + kp);

    int tile0 = (blockIdx.x * WAVES_PER_BLOCK + wave) * TILES_PER_WAVE;
    for (int t = 0; t < TILES_PER_WAVE; ++t) {
        int p = (tile0 + t) * 16 + ln;                    // this lane's point
        const float* F = feats + ((size_t)b * NPIX + p) * D8;
        v2f b_lo = *(const v2f*)(F + kp);                 // B: point column ln
        v2f b_hi = *(const v2f*)(F + 4 + kp);

        v8f acc0 = {}, acc1 = {};
        acc0 = wmma4(a0_lo, b_lo, acc0);
        acc0 = wmma4(a0_hi, b_hi, acc0);                  // dist, clusters 0..15
        acc1 = wmma4(a1_lo, b_lo, acc1);
        acc1 = wmma4(a1_hi, b_hi, acc1);                  // dist, clusters 16..31

        // in-lane argmin over this lane's 16 clusters (ascending order,
        // strict < keeps first occurrence => jnp.argmin tie semantics)
        float bd = acc0[0];
        int   bi = hi * 8;
        #pragma unroll
        for (int r = 1; r < 8; ++r) {
            float d = acc0[r];
            bool tk = d < bd;
            bd = tk ? d : bd;
            bi = tk ? (hi * 8 + r) : bi;
        }
        #pragma unroll
        for (int r = 0; r < 8; ++r) {
            float d = acc1[r];
            bool tk = d < bd;
            bd = tk ? d : bd;
            bi = tk ? (16 + hi * 8 + r) : bi;
        }
        // merge with partner half-wave (same point, other 16 clusters)
        float od = __int_as_float(
            __builtin_amdgcn_ds_swizzle(__float_as_int(bd), SWAPX16));
        int oi = __builtin_amdgcn_ds_swizzle(bi, SWAPX16);
        bool tk = (od < bd) || (od == bd && oi < bi);
        bd = tk ? od : bd;
        bi = tk ? oi : bi;

        if (hi == 0) {
            labels[(size_t)b * NPIX + p] = bi;            // 16 coalesced ints
            float4 f03 = *(const float4*)F;               // L0-hot reload
            float f4 = F[4];
            float* s = smem + bi * 6;
            atomicAdd(s + 0, f03.x);
            atomicAdd(s + 1, f03.y);
            atomicAdd(s + 2, f03.z);
            atomicAdd(s + 3, f03.w);
            atomicAdd(s + 4, f4);
            atomicAdd(s + 5, 1.0f);
        }
    }
    __syncthreads();
    if (tid < GS_STRIDE) atomicAdd(&gsums[b * GS_STRIDE + tid], smem[tid]);
}

// ---------------------------------------------------------------------------
// Kernel 4: centroid update c = sum / max(count,1), emitted in operand form
// {-2c, ||c||^2, 0, 0}; resets accumulators for the next assignment pass.
// ---------------------------------------------------------------------------
__global__ void update_kernel(float* __restrict__ gsums,
                              float* __restrict__ cop) {
    int i = threadIdx.x;            // 0..127 == flat (b,k)
    int b = i / KC, k = i % KC;
    float* s = gsums + b * GS_STRIDE + k * 6;
    float cnt = s[5];
    if (cnt < 1.0f) cnt = 1.0f;
    float inv = 1.0f / cnt;
    float nrm = 0.0f;
    float* cp = cop + (size_t)i * D8;
    #pragma unroll
    for (int j = 0; j < 5; ++j) {
        float v = s[j] * inv;
        cp[j] = -2.0f * v;
        nrm += v * v;
    }
    cp[5] = nrm; cp[6] = 0.0f; cp[7] = 0.0f;
    #pragma unroll
    for (int j = 0; j < 6; ++j) s[j] = 0.0f;
}

// ---------------------------------------------------------------------------
extern "C" void kernel_launch(void* const* d_in, const int* in_sizes, int n_in,
                              void* d_out, int out_size, void* d_ws, size_t ws_size,
                              hipStream_t stream) {
    (void)in_sizes; (void)n_in; (void)out_size; (void)ws_size;
    const float* x = (const float*)d_in[0];
    int* labels = (int*)d_out;                      // int32 labels [B,H,W]

    float* wsf   = (float*)d_ws;
    float* feats = wsf;                                         // B*N*8 f32
    float* cop   = feats + (size_t)BATCH * NPIX * D8;           // B*K*8
    float* gsums = cop + BATCH * KC * D8;                       // B*K*6

    feat_kernel<<<dim3(NPIX / 256, BATCH), 256, 0, stream>>>(x, feats);
    init_kernel<<<1, 128, 0, stream>>>(feats, cop, gsums);

    // labels_0 = assign(c_init); then 50x { c = update(prev labels); assign(c) }
    for (int it = 0; it <= 50; ++it) {
        assign_kernel<<<dim3(ASSIGN_BLOCKS_X, BATCH), 256, 0, stream>>>(
            feats, cop, gsums, labels);
        if (it < 50)
            update_kernel<<<1, 128, 0, stream>>>(gsums, cop);
    }
}